// ScaledDotProductAttention_28604482191373
// MI455X (gfx1250) — compile-verified
//
#include <hip/hip_runtime.h>
#include <hip/hip_bf16.h>

// ---------------------------------------------------------------------------
// Multi-scale "attention" for MI455X (gfx1250, wave32).
//
// Cost model (MI455X): conv phase = 34.7 G MACs fp32 -> done on the matrix
// pipe with V_WMMA_F32_16X16X4_F32 (exact fp32, matches the fp32 reference).
// Packed non-zero conv weights (~68 MB) are L2-resident (192 MB L2), so the
// GEMM is compute-bound, not HBM-bound (HBM touch ~200 MB -> ~10us @23.3TB/s).
// Softmax is over the *scale* axis (pointwise in q,k), so attn.sum(g) == 1.0
// identically -> attn_out output is a constant fill. Context phase (~2.7
// GFLOP) also uses fp32 WMMA with an LDS round-trip to convert the softmaxed
// score tile from C/D layout to A layout.
// ---------------------------------------------------------------------------

typedef float v2f __attribute__((ext_vector_type(2)));
typedef float v8f __attribute__((ext_vector_type(8)));

#define WMMA_F32X4(a, b, c)                                                    \
  __builtin_amdgcn_wmma_f32_16x16x4_f32(false, (a), false, (b), (short)0,      \
                                        (c), false, false)

#define B_ 2
#define H_ 8
#define L_ 512
#define DK_ 16
#define DM_ 128
#define NK_ 10
#define KMAX_ 512
#define KS_ 513
#define LP_ 1024  // L + KMAX

// K-dim offset (in reduction-index units) of scale g in the packed weights:
// koff(g) = 128 * sum_{g'<g} (2^g'+1) = 128 * ((1<<g)-1 + g)
__device__ __host__ inline int kofK(int g) { return 128 * ((1 << g) - 1 + g); }
#define KTOT 132224  // kofK(10) = 128*1033

// Workspace layout (float offsets)
#define XP_OFF 0u
#define XP_SZ (2u * 2u * 128u * 1024u)            // 524288
#define WP_OFF XP_SZ                              // 524288
#define WP_SZ ((unsigned)KTOT * 128u)             // 16924672
#define Y_OFF (WP_OFF + WP_SZ)                    // 17448960
#define Y_SZ (2u * 10u * 2u * 128u * 512u)        // 2621440

// ---------------------------------------------------------------------------
// Kernel 1: build left-padded, transposed inputs Xp[t][b][m][lp], lp in [0,1024)
// Xp = 0 for lp < 512, else Qf/Kf[b, lp-512, m]  (raw reshape of [B,H,L,DK])
// ---------------------------------------------------------------------------
__global__ void pack_x_kernel(const float* __restrict__ Q,
                              const float* __restrict__ K,
                              float* __restrict__ Xp) {
  unsigned i = blockIdx.x * 256u + threadIdx.x;  // 2^19 total
  int lp = i & 1023;
  int m = (i >> 10) & 127;
  int b = (i >> 17) & 1;
  int t = (i >> 18) & 1;
  const float* src = t ? K : Q;
  float v = 0.0f;
  if (lp >= KMAX_) v = src[(b * L_ + (lp - KMAX_)) * DM_ + m];
  Xp[i] = v;
}

// ---------------------------------------------------------------------------
// Kernel 2: pack non-zero conv taps into GEMM-friendly A^T layout
// Wp[g][k][co], k = j_tap*128 + ci  (co contiguous -> coalesced LDS fills)
// ---------------------------------------------------------------------------
__global__ void pack_w_kernel(const float* __restrict__ W,
                              float* __restrict__ Wp) {
  unsigned j = blockIdx.x * 256u + threadIdx.x;
  if (j >= (unsigned)KTOT * 128u) return;
  int co = j & 127;
  int kg = j >> 7;  // global K index in [0, KTOT)
  int g = 0;
  while (g < 9 && kg >= kofK(g + 1)) ++g;
  int kl = kg - kofK(g);
  int jt = kl >> 7;
  int ci = kl & 127;
  int klen = (1 << g) + 1;
  size_t src = ((size_t)(g * 128 + co) * 128 + ci) * (size_t)KS_ +
               (size_t)(KS_ - klen) + jt;
  Wp[j] = W[src];
}

// ---------------------------------------------------------------------------
// Kernel 3: conv as GEMM with fp32 WMMA.
// Per block (t,g,b, nblk): Y tile [128 co x 64 l] = Wp_g(128 x Kg) @ Xcol.
// 8 waves; wave w owns rows 16w..16w+15, 4 accum tiles of 16x16.
// ---------------------------------------------------------------------------
__global__ __launch_bounds__(256) void conv_gemm_kernel(
    const float* __restrict__ Wp, const float* __restrict__ Xp,
    const float* __restrict__ bconv, float* __restrict__ Y) {
  __shared__ float Alds[16 * 128];  // [k][co]
  __shared__ float Blds[16 * 64];   // [k][l]

  int z = blockIdx.z;         // (t*10+g)*2 + b
  int b = z & 1;
  int g = (z >> 1) % 10;
  int t = z / 20;
  int klen = (1 << g) + 1;
  int Kg = 128 * klen;
  int shiftL = KS_ - klen;  // = KMAX+1-klen

  const float* Wg = Wp + (size_t)kofK(g) * 128;
  const float* Xb = Xp + (size_t)((t * 2 + b) * 128) * LP_;
  int l0 = blockIdx.x * 64;

  int tid = threadIdx.x;
  int wave = tid >> 5;
  int lane = tid & 31;
  int lo = lane & 15;
  int hi = lane >> 4;

  v8f acc[4];
#pragma unroll
  for (int nt = 0; nt < 4; ++nt) acc[nt] = (v8f){0, 0, 0, 0, 0, 0, 0, 0};

  for (int k0 = 0; k0 < Kg; k0 += 16) {
    // --- cooperative fills ---
    {
      int rowA = tid >> 4;
      int colA = (tid & 15) * 8;
      const float4* sA =
          reinterpret_cast<const float4*>(Wg + (size_t)(k0 + rowA) * 128 + colA);
      float4 a0 = sA[0];
      float4 a1 = sA[1];
      *reinterpret_cast<float4*>(&Alds[rowA * 128 + colA]) = a0;
      *reinterpret_cast<float4*>(&Alds[rowA * 128 + colA + 4]) = a1;

      int rowB = tid >> 4;
      int colB = (tid & 15) * 4;
      int kk = k0 + rowB;
      int ci = kk & 127;
      int jt = kk >> 7;
      const float* sB = Xb + (size_t)ci * LP_ + (l0 + shiftL + jt);
#pragma unroll
      for (int c = 0; c < 4; ++c) Blds[rowB * 64 + colB + c] = sB[colB + c];
    }
    __syncthreads();

    // --- 16 WMMAs per wave per chunk ---
#pragma unroll
    for (int ks = 0; ks < 4; ++ks) {
      int kr = ks * 4 + 2 * hi;
      int cw = wave * 16 + lo;
      v2f a;
      a.x = Alds[kr * 128 + cw];
      a.y = Alds[(kr + 1) * 128 + cw];
#pragma unroll
      for (int nt = 0; nt < 4; ++nt) {
        v2f bb;
        bb.x = Blds[kr * 64 + nt * 16 + lo];
        bb.y = Blds[(kr + 1) * 64 + nt * 16 + lo];
        acc[nt] = WMMA_F32X4(a, bb, acc[nt]);
      }
    }
    __syncthreads();
  }

  // epilogue: + bias, store Y[t][g][b][co][l]
  size_t ybase = ((size_t)z) * 128 * 512;  // z == (t*10+g)*2+b
#pragma unroll
  for (int r = 0; r < 8; ++r) {
    int mm = wave * 16 + r + 8 * hi;
    float bv = bconv[g * 128 + mm];
    size_t rowo = ybase + (size_t)mm * 512 + l0;
#pragma unroll
    for (int nt = 0; nt < 4; ++nt) Y[rowo + nt * 16 + lo] = acc[nt][r] + bv;
  }
}

// ---------------------------------------------------------------------------
// Kernel 4: attn_out = softmax(scores, axis=g).sum(axis=g) == 1.0 exactly.
// ---------------------------------------------------------------------------
__global__ void fill_ones_kernel(float4* __restrict__ out) {
  unsigned i = blockIdx.x * 256u + threadIdx.x;  // 1048576 float4 = 4194304 f
  out[i] = make_float4(1.f, 1.f, 1.f, 1.f);
}

// ---------------------------------------------------------------------------
// Kernel 5: context[b,h,q,d] = sum_{g,k} softmax_g(QK^T/4 masked) * K_p
// One wave per (b,h,qtile16); streams over 32 k-tiles; softmax over g is
// pointwise. Score tiles kept in 10 v8f accumulators; attn tile goes through
// LDS to convert C/D layout -> A layout for the context WMMAs.
// Q_p[b,h,g,q,d] = Y[0][g][b][h*16+(q>>5)][(q&31)*16+d]  (raw-reshape map)
// ---------------------------------------------------------------------------
__global__ __launch_bounds__(128) void context_kernel(
    const float* __restrict__ Y, float* __restrict__ outCtx) {
  __shared__ float att[4][16 * 16];  // per-wave [m][k] tile

  int lane = threadIdx.x & 31;
  int wave = threadIdx.x >> 5;
  int lo = lane & 15;
  int hi = lane >> 4;

  int wg = blockIdx.x * 4 + wave;  // 512 waves total: [b][h][qt]
  int b = wg >> 8;
  int h = (wg >> 5) & 7;
  int qt = wg & 31;

  const float* Yq = Y;                                 // t = 0 (Q_p)
  const float* Yk = Y + (size_t)10 * 2 * 128 * 512;    // t = 1 (K_p)

  int q0 = qt * 16;
  int coq = h * 16 + (qt >> 1);
  int qoff = (qt & 1) * 16;

  v8f ctx = (v8f){0, 0, 0, 0, 0, 0, 0, 0};
  float* at = att[wave];

  for (int kt = 0; kt < 32; ++kt) {
    int k0 = kt * 16;
    int cok = h * 16 + (kt >> 1);
    int koff = (kt & 1) * 16;

    v8f s[10];
    // ---- scores: s_g = (Q_p tile) @ (K_p tile)^T * 0.25 ----
#pragma unroll
    for (int g = 0; g < 10; ++g) {
      const float* Aq =
          Yq + (((size_t)g * 2 + b) * 128 + coq) * 512 + (size_t)qoff * 16;
      const float* Bk =
          Yk + (((size_t)g * 2 + b) * 128 + cok) * 512 + (size_t)koff * 16;
      v8f c = (v8f){0, 0, 0, 0, 0, 0, 0, 0};
#pragma unroll
      for (int dc = 0; dc < 4; ++dc) {
        int dr = dc * 4 + 2 * hi;
        v2f a, bb;
        a.x = Aq[lo * 16 + dr];       // A[m=lo, dr]
        a.y = Aq[lo * 16 + dr + 1];
        bb.x = Bk[lo * 16 + dr];      // B[dr, n=lo] = K_p[k0+lo, dr]
        bb.y = Bk[lo * 16 + dr + 1];
        c = WMMA_F32X4(a, bb, c);
      }
      s[g] = c * 0.25f;
    }

    // ---- mask (k > q -> -1e9) ----
#pragma unroll
    for (int r = 0; r < 8; ++r) {
      if ((k0 + lo) > (q0 + r + 8 * hi)) {
#pragma unroll
        for (int g = 0; g < 10; ++g) s[g][r] = -1e9f;
      }
    }

    // ---- softmax over the 10 scales (pointwise per (q,k)) ----
#pragma unroll
    for (int r = 0; r < 8; ++r) {
      float mx = s[0][r];
#pragma unroll
      for (int g = 1; g < 10; ++g) mx = fmaxf(mx, s[g][r]);
      float sum = 0.f;
      float e[10];
#pragma unroll
      for (int g = 0; g < 10; ++g) {
        e[g] = __expf(s[g][r] - mx);
        sum += e[g];
      }
      float inv = 1.0f / sum;
#pragma unroll
      for (int g = 0; g < 10; ++g) s[g][r] = e[g] * inv;
    }

    // ---- context accumulation: ctx += attn_g @ K_p_g ----
#pragma unroll
    for (int g = 0; g < 10; ++g) {
      // spill attn tile (C/D layout) to LDS as [m][k]; in-order LDS within a
      // wave makes the transpose read-back safe without a barrier.
#pragma unroll
      for (int r = 0; r < 8; ++r) at[(r + 8 * hi) * 16 + lo] = s[g][r];
      const float* Bk =
          Yk + (((size_t)g * 2 + b) * 128 + cok) * 512 + (size_t)koff * 16;
#pragma unroll
      for (int kc = 0; kc < 4; ++kc) {
        int kr = kc * 4 + 2 * hi;
        v2f a, bb;
        a.x = at[lo * 16 + kr];        // A[m=lo, kr] (attn)
        a.y = at[lo * 16 + kr + 1];
        bb.x = Bk[kr * 16 + lo];       // B[kr, n=d=lo] = K_p[k0+kr, lo]
        bb.y = Bk[(kr + 1) * 16 + lo];
        ctx = WMMA_F32X4(a, bb, ctx);
      }
    }
  }

  // store context[b][h][q][d]
#pragma unroll
  for (int r = 0; r < 8; ++r) {
    int q = q0 + r + 8 * hi;
    outCtx[(((size_t)(b * 8 + h) * 512) + q) * 16 + lo] = ctx[r];
  }
}

// ---------------------------------------------------------------------------
extern "C" void kernel_launch(void* const* d_in, const int* in_sizes, int n_in,
                              void* d_out, int out_size, void* d_ws,
                              size_t ws_size, hipStream_t stream) {
  const float* Q = (const float*)d_in[0];
  const float* K = (const float*)d_in[1];
  // d_in[2] = V (unused by the 'con' branch)
  const float* W = (const float*)d_in[3];
  const float* bconv = (const float*)d_in[4];
  // d_in[5] = attn_mask (reproduced analytically)

  float* ws = (float*)d_ws;
  float* Xp = ws + XP_OFF;
  float* Wp = ws + WP_OFF;
  float* Y = ws + Y_OFF;  // needs ~80.3 MB of workspace total

  float* outCtx = (float*)d_out;                 // [B,H,L,DK] = 131072 floats
  float* outAttn = outCtx + (size_t)B_ * H_ * L_ * DK_;  // [B,H,L,L] = 1.0f

  pack_x_kernel<<<2048, 256, 0, stream>>>(Q, K, Xp);
  pack_w_kernel<<<(KTOT * 128 + 255) / 256, 256, 0, stream>>>(W, Wp);
  conv_gemm_kernel<<<dim3(8, 1, 40), 256, 0, stream>>>(Wp, Xp, bconv, Y);
  fill_ones_kernel<<<4096, 256, 0, stream>>>((float4*)outAttn);
  context_kernel<<<128, 128, 0, stream>>>(Y, outCtx);
}